// GraphNet_39848706572469
// MI455X (gfx1250) — compile-verified
//
#include <hip/hip_runtime.h>
#include <math.h>

// ---------------------------------------------------------------------------
// Problem constants (from the reference)
// ---------------------------------------------------------------------------
#define NNODES 25600      // N = B * NPG
#define NBATCH 64         // B
#define NPG    400        // nodes per graph
#define DIM    256        // D
#define NH     4          // heads
#define DHD    64         // D / H
#define HIDN   512        // HID
#define NEDGE  204800     // E = N * DEG
#define NEGF   (-1e30f)
#define EPSF   (1e-5f)

#define CDIV(a,b) (((a)+(b)-1)/(b))

typedef float v2f __attribute__((ext_vector_type(2)));
typedef float v8f __attribute__((ext_vector_type(8)));

// ---------------------------------------------------------------------------
// fp32 WMMA GEMM:  C[M,Nc] = A[M,K] @ W[K,Nc] (+bias) (optional relu)
//
// Block = 256 threads (8 waves). Block output panel = 128 rows x 64 cols.
//  - Each wave owns a 16-row strip and 4 accumulators (16x64), so one A
//    fragment feeds 4 WMMAs (4x A-load reuse in registers).
//  - The 64-col weight panel is staged into LDS in K-chunks of 128 via
//    GLOBAL_LOAD_ASYNC_TO_LDS_B128 (ASYNCcnt / s_wait_asynccnt), shared by
//    all 8 waves (8x less W global traffic).
//  - LDS rows padded to 72 floats: bank(elem) = (8*r + c) % 64, so the two
//    half-wave row reads (r and r+2) land on disjoint 16-bank sets.
// Fragment layouts per CDNA5 ISA 7.12.2:
//   A 16x4 : lanes 0-15 row=lane, VGPR0=K+0, VGPR1=K+1; lanes16-31: K+2,K+3
//   B 4x16 : mirrored (row K striped across lanes within a VGPR)
//   C 16x16: vgpr v, lanes0-15 -> M=v, lanes16-31 -> M=v+8, N=lane%16
// ---------------------------------------------------------------------------
#define KCHUNK 128
#define LDSPAD 72

__global__ __launch_bounds__(256)
void gemm16(const float* __restrict__ A, const float* __restrict__ W,
            const float* __restrict__ bias, float* __restrict__ C,
            int M, int K, int Nc, int relu) {
  __shared__ float tile[KCHUNK * LDSPAD];   // 36 KB

  const int tid  = threadIdx.x;
  const int wave = tid >> 5;
  const int lane = tid & 31;
  const int half = lane >> 4;
  const int lm   = lane & 15;
  const int col0 = blockIdx.x * 64;
  const int row0 = blockIdx.y * 128 + wave * 16;
  const bool rowact = (row0 < M);           // wave-uniform

  v8f acc[4];
  #pragma unroll
  for (int t = 0; t < 4; ++t) acc[t] = (v8f){0.f,0.f,0.f,0.f,0.f,0.f,0.f,0.f};

  for (int kc = 0; kc < K; kc += KCHUNK) {
    // ---- cooperative async stage of W[kc..kc+128) x cols[col0..col0+64)
    #pragma unroll
    for (int it = 0; it < (KCHUNK * 64) / (256 * 4); ++it) {   // 8 iters
      int g = (it * 256 + tid) * 4;
      int r = g >> 6;            // 0..127
      int c = g & 63;
      unsigned lds = (unsigned)(uintptr_t)(&tile[r * LDSPAD + c]);
      const float* gp = W + (size_t)(kc + r) * Nc + col0 + c;
      asm volatile("global_load_async_to_lds_b128 %0, %1, off"
                   :: "v"(lds), "v"((unsigned long long)(uintptr_t)gp)
                   : "memory");
    }
    asm volatile("s_wait_asynccnt 0x0" ::: "memory");
    __syncthreads();

    // ---- consume chunk
    if (rowact) {
      const float* arow = A + (size_t)(row0 + lm) * K + kc;
      for (int k = 0; k < KCHUNK; k += 4) {
        v2f a;
        a.x = arow[k + 2*half];
        a.y = arow[k + 2*half + 1];
        #pragma unroll
        for (int t = 0; t < 4; ++t) {
          v2f b;
          b.x = tile[(k + 2*half)     * LDSPAD + t*16 + lm];
          b.y = tile[(k + 2*half + 1) * LDSPAD + t*16 + lm];
          acc[t] = __builtin_amdgcn_wmma_f32_16x16x4_f32(false, a, false, b,
                                                         (short)0, acc[t],
                                                         false, false);
        }
      }
    }
    __syncthreads();
  }

  if (rowact) {
    #pragma unroll
    for (int t = 0; t < 4; ++t) {
      const int col = col0 + t*16 + lm;
      const float bv = bias ? bias[col] : 0.f;
      #pragma unroll
      for (int v = 0; v < 8; ++v) {
        int row = row0 + v + 8*half;
        float val = acc[t][v] + bv;
        if (relu) val = fmaxf(val, 0.f);
        C[(size_t)row * Nc + col] = val;
      }
    }
  }
}

// ---------------------------------------------------------------------------
// small helpers
// ---------------------------------------------------------------------------
__device__ __forceinline__ unsigned f2ord(float f) {
  unsigned u = __float_as_uint(f);
  return (u & 0x80000000u) ? ~u : (u | 0x80000000u);
}
__device__ __forceinline__ float ord2f(unsigned u) {
  unsigned v = (u & 0x80000000u) ? (u & 0x7FFFFFFFu) : ~u;
  return __uint_as_float(v);
}

__global__ void fill_f32(float* p, float v, int n) {
  int i = blockIdx.x * blockDim.x + threadIdx.x;
  if (i < n) p[i] = v;
}
__global__ void fill_i32(int* p, int v, int n) {
  int i = blockIdx.x * blockDim.x + threadIdx.x;
  if (i < n) p[i] = v;
}
__global__ void fill_ord(unsigned* p, int n) {
  int i = blockIdx.x * blockDim.x + threadIdx.x;
  if (i < n) p[i] = f2ord(NEGF);
}

// ---------------------------------------------------------------------------
// GCN pieces
// ---------------------------------------------------------------------------
__global__ void deg_scatter(const int* __restrict__ dst, float* __restrict__ deg) {
  int e = blockIdx.x * blockDim.x + threadIdx.x;
  if (e < NEDGE) atomicAdd(&deg[dst[e]], 1.0f);
}
// agg init: out[n,d] = s[n,d]/deg[n] + b[d]
__global__ void gcn_self(const float* __restrict__ s, const float* __restrict__ deg,
                         const float* __restrict__ b, float* __restrict__ out) {
  int i = blockIdx.x * blockDim.x + threadIdx.x;
  if (i >= NNODES * DIM) return;
  int n = i >> 8, d = i & 255;
  out[i] = s[i] / deg[n] + b[d];
}
// edge scatter: agg[dst] += s[src] * rsqrt(deg[src])*rsqrt(deg[dst])
__global__ void gcn_scatter(const float* __restrict__ s, const int* __restrict__ src,
                            const int* __restrict__ dst, const float* __restrict__ deg,
                            float* __restrict__ agg) {
  int idx = blockIdx.x * blockDim.x + threadIdx.x;
  if (idx >= NEDGE * (DIM/4)) return;
  int e = idx >> 6, c = idx & 63;
  int is = src[e], id = dst[e];
  float coef = rsqrtf(deg[is]) * rsqrtf(deg[id]);
  const float4 v = *(const float4*)(s + (size_t)is * DIM + c*4);
  float* o = agg + (size_t)id * DIM + c*4;
  atomicAdd(o+0, v.x*coef); atomicAdd(o+1, v.y*coef);
  atomicAdd(o+2, v.z*coef); atomicAdd(o+3, v.w*coef);
}
__global__ void relu_k(const float* __restrict__ a, float* __restrict__ o, int n) {
  int i = blockIdx.x * blockDim.x + threadIdx.x;
  if (i < n) o[i] = fmaxf(a[i], 0.f);
}
__global__ void logsoftmax_k(const float* __restrict__ a, float* __restrict__ o) {
  int n = blockIdx.x, t = threadIdx.x;
  __shared__ float red[DIM];
  size_t base = (size_t)n * DIM;
  float x = a[base + t];
  red[t] = x; __syncthreads();
  for (int s = 128; s > 0; s >>= 1) { if (t < s) red[t] = fmaxf(red[t], red[t+s]); __syncthreads(); }
  float m = red[0]; __syncthreads();
  red[t] = expf(x - m); __syncthreads();
  for (int s = 128; s > 0; s >>= 1) { if (t < s) red[t] += red[t+s]; __syncthreads(); }
  o[base + t] = x - m - logf(red[0]);
}

// ---------------------------------------------------------------------------
// Transformer attention pieces
// ---------------------------------------------------------------------------
__global__ void attn_logit(const float* __restrict__ q, const float* __restrict__ k,
                           const int* __restrict__ src, const int* __restrict__ dst,
                           const int* __restrict__ emask, float* __restrict__ logit) {
  int idx = blockIdx.x * blockDim.x + threadIdx.x;
  if (idx >= NEDGE * NH) return;
  int e = idx >> 2, h = idx & 3;
  if (!emask[e]) { logit[idx] = NEGF; return; }
  const float4* qr = (const float4*)(q + (size_t)dst[e] * DIM + h * DHD);
  const float4* kr = (const float4*)(k + (size_t)src[e] * DIM + h * DHD);
  float acc = 0.f;
  #pragma unroll
  for (int j = 0; j < DHD/4; ++j) {
    float4 a = qr[j], b = kr[j];
    acc += a.x*b.x + a.y*b.y + a.z*b.z + a.w*b.w;
  }
  logit[idx] = acc * 0.125f;  // 1/sqrt(64)
}
__global__ void seg_max(const float* __restrict__ logit, const int* __restrict__ dst,
                        unsigned* __restrict__ mord) {
  int idx = blockIdx.x * blockDim.x + threadIdx.x;
  if (idx >= NEDGE * NH) return;
  int e = idx >> 2, h = idx & 3;
  atomicMax(&mord[dst[e] * NH + h], f2ord(logit[idx]));
}
__global__ void pe_den(const float* __restrict__ logit, const int* __restrict__ dst,
                       const int* __restrict__ emask, const unsigned* __restrict__ mord,
                       float* __restrict__ pe, float* __restrict__ den) {
  int idx = blockIdx.x * blockDim.x + threadIdx.x;
  if (idx >= NEDGE * NH) return;
  int e = idx >> 2, h = idx & 3;
  float p = 0.f;
  if (emask[e]) p = expf(logit[idx] - ord2f(mord[dst[e] * NH + h]));
  pe[idx] = p;
  if (p != 0.f) atomicAdd(&den[dst[e] * NH + h], p);
}
__global__ void msg_scatter(const float* __restrict__ pe, const float* __restrict__ v,
                            const int* __restrict__ src, const int* __restrict__ dst,
                            float* __restrict__ msg) {
  int idx = blockIdx.x * blockDim.x + threadIdx.x;
  if (idx >= NEDGE * (DIM/4)) return;
  int e = idx >> 6, c = idx & 63;       // c*4 = feature index, head = c>>4
  float p = pe[e * NH + (c >> 4)];
  if (p == 0.f) return;
  const float4 vv = *(const float4*)(v + (size_t)src[e] * DIM + c*4);
  float* o = msg + (size_t)dst[e] * DIM + c*4;
  atomicAdd(o+0, p*vv.x); atomicAdd(o+1, p*vv.y);
  atomicAdd(o+2, p*vv.z); atomicAdd(o+3, p*vv.w);
}
// o_t = msg/max(den,..) + skip  (written in-place into skip)
__global__ void attn_finish(const float* __restrict__ msg, const float* __restrict__ den,
                            float* __restrict__ skipo) {
  int i = blockIdx.x * blockDim.x + threadIdx.x;
  if (i >= NNODES * DIM) return;
  int n = i >> 8, d = i & 255;
  float dn = den[n * NH + (d >> 6)];
  skipo[i] = msg[i] / (dn > 0.f ? dn : 1.f) + skipo[i];
}
// out = LayerNorm(a + b) * g + beta   (row length DIM)
__global__ void add_ln(const float* __restrict__ a, const float* __restrict__ b,
                       const float* __restrict__ g, const float* __restrict__ beta,
                       float* __restrict__ out) {
  int n = blockIdx.x, t = threadIdx.x;
  __shared__ float red[DIM];
  size_t base = (size_t)n * DIM;
  float x = a[base + t] + b[base + t];
  red[t] = x; __syncthreads();
  for (int s = 128; s > 0; s >>= 1) { if (t < s) red[t] += red[t+s]; __syncthreads(); }
  float mu = red[0] * (1.f / DIM); __syncthreads();
  float dx = x - mu;
  red[t] = dx * dx; __syncthreads();
  for (int s = 128; s > 0; s >>= 1) { if (t < s) red[t] += red[t+s]; __syncthreads(); }
  float var = red[0] * (1.f / DIM);
  out[base + t] = dx * rsqrtf(var + EPSF) * g[t] + beta[t];
}

// ---------------------------------------------------------------------------
// SAG pooling pieces
// ---------------------------------------------------------------------------
__global__ void sag_scatter(const float* __restrict__ h, const int* __restrict__ src,
                            const int* __restrict__ dst, const int* __restrict__ emask,
                            float* __restrict__ agg) {
  int idx = blockIdx.x * blockDim.x + threadIdx.x;
  if (idx >= NEDGE * (DIM/4)) return;
  int e = idx >> 6, c = idx & 63;
  if (!emask[e]) return;
  const float4 v = *(const float4*)(h + (size_t)src[e] * DIM + c*4);
  float* o = agg + (size_t)dst[e] * DIM + c*4;
  atomicAdd(o+0, v.x); atomicAdd(o+1, v.y); atomicAdd(o+2, v.z); atomicAdd(o+3, v.w);
}
__global__ void score_k(const float* __restrict__ agg, const float* __restrict__ h,
                        const float* __restrict__ relw, const float* __restrict__ relb,
                        const float* __restrict__ rootw, float* __restrict__ score) {
  int n = blockIdx.x * blockDim.x + threadIdx.x;
  if (n >= NNODES) return;
  const float* ar = agg + (size_t)n * DIM;
  const float* hr = h   + (size_t)n * DIM;
  float s = 0.f;
  for (int j = 0; j < DIM; ++j) s += ar[j]*relw[j] + hr[j]*rootw[j];
  score[n] = s + relb[0];
}
// Exact stable top-K per graph via rank counting (ties -> lower index first,
// matching lax.top_k). One block per graph.
__global__ void topk_k(const float* __restrict__ score, const int* __restrict__ nmask,
                       int K, int* __restrict__ nm) {
  int g = blockIdx.x, t = threadIdx.x;
  __shared__ float sc[NPG];
  if (t < NPG) {
    int n = g * NPG + t;
    sc[t] = nmask[n] ? score[n] : NEGF;
  }
  __syncthreads();
  if (t < NPG) {
    float si = sc[t];
    int r = 0;
    for (int j = 0; j < NPG; ++j) {
      float sj = sc[j];
      r += (sj > si) || (sj == si && j < t);
    }
    nm[g * NPG + t] = (r < K) ? 1 : 0;
  }
}
__global__ void emask_update(const int* __restrict__ src, const int* __restrict__ dst,
                             const int* __restrict__ nm, int* __restrict__ emask) {
  int e = blockIdx.x * blockDim.x + threadIdx.x;
  if (e < NEDGE) emask[e] = emask[e] && nm[src[e]] && nm[dst[e]];
}
__global__ void update_h(float* __restrict__ h, const float* __restrict__ score,
                         const int* __restrict__ nm, int* __restrict__ nmask) {
  int i = blockIdx.x * blockDim.x + threadIdx.x;
  if (i >= NNODES * DIM) return;
  int n = i >> 8, d = i & 255;
  h[i] *= tanhf(score[n]) * (nm[n] ? 1.f : 0.f);
  if (d == 0) nmask[n] = nm[n];
}
__global__ void readout_k(const float* __restrict__ h, const int* __restrict__ nmask,
                          float* __restrict__ z, float kk) {
  int g = blockIdx.x, d = threadIdx.x;
  float mx = NEGF, sm = 0.f;
  for (int i = 0; i < NPG; ++i) {
    int n = g * NPG + i;
    if (nmask[n]) {
      float v = h[(size_t)n * DIM + d];
      mx = fmaxf(mx, v);
      sm += v;
    }
  }
  z[(size_t)g * 2 * DIM + d]       += mx;
  z[(size_t)g * 2 * DIM + DIM + d] += sm / kk;
}

// ---------------------------------------------------------------------------
// Head pieces
// ---------------------------------------------------------------------------
__global__ void bn_relu(float* __restrict__ z, const float* __restrict__ g,
                        const float* __restrict__ b, int M, int C) {
  int c = threadIdx.x;
  if (c >= C) return;
  float s = 0.f;
  for (int r = 0; r < M; ++r) s += z[r*C + c];
  float mu = s / M;
  float v = 0.f;
  for (int r = 0; r < M; ++r) { float d = z[r*C + c] - mu; v += d*d; }
  float sc = rsqrtf(v / M + EPSF) * g[c];
  float bb = b[c];
  for (int r = 0; r < M; ++r) z[r*C + c] = fmaxf((z[r*C + c] - mu) * sc + bb, 0.f);
}
__global__ void final_k(const float* __restrict__ z2, const float* __restrict__ w,
                        const float* __restrict__ b, float* __restrict__ out) {
  int t = threadIdx.x;
  if (t >= NBATCH) return;
  float s = 0.f;
  for (int j = 0; j < DIM/2; ++j) s += z2[t * (DIM/2) + j] * w[j];
  out[t] = s + b[0];
}

// ---------------------------------------------------------------------------
// Host-side orchestration
// ---------------------------------------------------------------------------
extern "C" void kernel_launch(void* const* d_in, const int* in_sizes, int n_in,
                              void* d_out, int out_size, void* d_ws, size_t ws_size,
                              hipStream_t stream) {
  (void)in_sizes; (void)n_in; (void)out_size; (void)ws_size;

  // --- param leaf indices (insertion-order recursive flatten of setup_inputs)
  const float* x    = (const float*)d_in[0];
  const int*   src  = (const int*)d_in[1];
  const int*   dst  = src + NEDGE;
  // d_in[2] = batch (layout is structured; unused)
  const float* gc1w = (const float*)d_in[3];
  const float* gc1b = (const float*)d_in[4];
  const float* gc2w = (const float*)d_in[5];
  const float* gc2b = (const float*)d_in[6];
  #define GT(i, off)   ((const float*)d_in[7 + 18*(i) + (off)])
  #define POOL(i, off) ((const float*)d_in[61 + 3*(i) + (off)])
  #define HD(off)      ((const float*)d_in[70 + (off)])
  float* out = (float*)d_out;

  // --- workspace carve-out (~218 MB)
  char* cur = (char*)d_ws;
  auto alloc = [&](size_t bytes) -> void* {
    void* r = (void*)cur;
    cur += (bytes + 255) & ~(size_t)255;
    return r;
  };
  const size_t ND = (size_t)NNODES * DIM;
  float*    w_h    = (float*)alloc(ND * 4);
  float*    w_q    = (float*)alloc(ND * 4);
  float*    w_k    = (float*)alloc(ND * 4);
  float*    w_v    = (float*)alloc(ND * 4);
  float*    w_skip = (float*)alloc(ND * 4);
  float*    w_msg  = (float*)alloc(ND * 4);
  float*    w_hid  = (float*)alloc((size_t)NNODES * HIDN * 4);
  float*    w_log  = (float*)alloc((size_t)NEDGE * NH * 4);
  float*    w_pe   = (float*)alloc((size_t)NEDGE * NH * 4);
  unsigned* w_mord = (unsigned*)alloc((size_t)NNODES * NH * 4);
  float*    w_den  = (float*)alloc((size_t)NNODES * NH * 4);
  float*    w_deg  = (float*)alloc((size_t)NNODES * 4);
  float*    w_scr  = (float*)alloc((size_t)NNODES * 4);
  float*    w_z    = (float*)alloc((size_t)NBATCH * 2 * DIM * 4);
  float*    w_z1   = (float*)alloc((size_t)NBATCH * DIM * 4);
  float*    w_z2   = (float*)alloc((size_t)NBATCH * (DIM/2) * 4);
  int*      w_em   = (int*)alloc((size_t)NEDGE * 4);
  int*      w_nm   = (int*)alloc((size_t)NNODES * 4);
  int*      w_nmk  = (int*)alloc((size_t)NNODES * 4);

  const dim3 B256(256);
  const int gND  = CDIV((int)ND, 256);
  const int gEH  = CDIV(NEDGE * NH, 256);
  const int gE64 = CDIV(NEDGE * (DIM/4), 256);
  const int gE   = CDIV(NEDGE, 256);
  // gemm grids: (cols/64, rows/128), 256 threads
  const dim3 gmmDD(DIM/64,  CDIV(NNODES,128));   // [N,*]->[N,256]
  const dim3 gmmDH(HIDN/64, CDIV(NNODES,128));   // [N,256]->[N,512]

  // === init =================================================================
  fill_f32<<<CDIV(NNODES,256), B256, 0, stream>>>(w_deg, 1.0f, NNODES);
  deg_scatter<<<gE, B256, 0, stream>>>(dst, w_deg);
  fill_f32<<<CDIV(NBATCH*2*DIM,256), B256, 0, stream>>>(w_z, 0.f, NBATCH*2*DIM);
  fill_i32<<<gE, B256, 0, stream>>>(w_em, 1, NEDGE);
  fill_i32<<<CDIV(NNODES,256), B256, 0, stream>>>(w_nmk, 1, NNODES);

  // === GCN block ============================================================
  // h = relu(gcn1(x))
  gemm16<<<gmmDD, B256, 0, stream>>>(x, gc1w, nullptr, w_q, NNODES, DIM, DIM, 0);
  gcn_self<<<gND, B256, 0, stream>>>(w_q, w_deg, gc1b, w_k);
  gcn_scatter<<<gE64, B256, 0, stream>>>(w_q, src, dst, w_deg, w_k);
  relu_k<<<gND, B256, 0, stream>>>(w_k, w_h, (int)ND);
  // h = log_softmax(gcn2(h))
  gemm16<<<gmmDD, B256, 0, stream>>>(w_h, gc2w, nullptr, w_q, NNODES, DIM, DIM, 0);
  gcn_self<<<gND, B256, 0, stream>>>(w_q, w_deg, gc2b, w_k);
  gcn_scatter<<<gE64, B256, 0, stream>>>(w_q, src, dst, w_deg, w_k);
  logsoftmax_k<<<NNODES, DIM, 0, stream>>>(w_k, w_h);

  // === 3 x (graph transformer -> SAG pool -> readout) =======================
  const int   Ks[3]  = {320, 256, 205};
  const float kkf[3] = {320.f, 256.f, 205.f};
  for (int L = 0; L < 3; ++L) {
    // ---- tconv projections
    gemm16<<<gmmDD, B256, 0, stream>>>(w_h, GT(L,0),  GT(L,1), w_q,    NNODES, DIM, DIM, 0);
    gemm16<<<gmmDD, B256, 0, stream>>>(w_h, GT(L,2),  GT(L,3), w_k,    NNODES, DIM, DIM, 0);
    gemm16<<<gmmDD, B256, 0, stream>>>(w_h, GT(L,4),  GT(L,5), w_v,    NNODES, DIM, DIM, 0);
    gemm16<<<gmmDD, B256, 0, stream>>>(w_h, GT(L,6),  GT(L,7), w_skip, NNODES, DIM, DIM, 0);
    // ---- edge-softmax attention
    attn_logit<<<gEH, B256, 0, stream>>>(w_q, w_k, src, dst, w_em, w_log);
    fill_ord<<<CDIV(NNODES*NH,256), B256, 0, stream>>>(w_mord, NNODES*NH);
    seg_max<<<gEH, B256, 0, stream>>>(w_log, dst, w_mord);
    fill_f32<<<CDIV(NNODES*NH,256), B256, 0, stream>>>(w_den, 0.f, NNODES*NH);
    pe_den<<<gEH, B256, 0, stream>>>(w_log, dst, w_em, w_mord, w_pe, w_den);
    fill_f32<<<gND, B256, 0, stream>>>(w_msg, 0.f, (int)ND);
    msg_scatter<<<gE64, B256, 0, stream>>>(w_pe, w_v, src, dst, w_msg);
    attn_finish<<<gND, B256, 0, stream>>>(w_msg, w_den, w_skip);   // o_t -> w_skip
    // ---- out proj + LN + FFN + LN
    gemm16<<<gmmDD, B256, 0, stream>>>(w_skip, GT(L,8), GT(L,9), w_k, NNODES, DIM, DIM, 0); // o1
    add_ln<<<NNODES, DIM, 0, stream>>>(w_k, w_h, GT(L,10), GT(L,11), w_v);                   // o2
    gemm16<<<gmmDH, B256, 0, stream>>>(w_v, GT(L,12), GT(L,13), w_hid, NNODES, DIM, HIDN, 1);
    gemm16<<<gmmDD, B256, 0, stream>>>(w_hid, GT(L,14), GT(L,15), w_q, NNODES, HIDN, DIM, 0); // o3
    add_ln<<<NNODES, DIM, 0, stream>>>(w_q, w_v, GT(L,16), GT(L,17), w_h);                   // new h
    // ---- SAG pooling
    fill_f32<<<gND, B256, 0, stream>>>(w_msg, 0.f, (int)ND);
    sag_scatter<<<gE64, B256, 0, stream>>>(w_h, src, dst, w_em, w_msg);
    score_k<<<CDIV(NNODES,256), B256, 0, stream>>>(w_msg, w_h, POOL(L,0), POOL(L,1),
                                                   POOL(L,2), w_scr);
    topk_k<<<NBATCH, 512, 0, stream>>>(w_scr, w_nmk, Ks[L], w_nm);
    emask_update<<<gE, B256, 0, stream>>>(src, dst, w_nm, w_em);
    update_h<<<gND, B256, 0, stream>>>(w_h, w_scr, w_nm, w_nmk);   // also nmask <- nm
    // ---- readout (accumulate x1+x2+x3 into w_z)
    readout_k<<<NBATCH, DIM, 0, stream>>>(w_h, w_nmk, w_z, kkf[L]);
  }

  // === head =================================================================
  gemm16<<<dim3(DIM/64, 1), B256, 0, stream>>>(w_z, HD(0), HD(1), w_z1,
                                               NBATCH, 2*DIM, DIM, 0);
  bn_relu<<<1, DIM, 0, stream>>>(w_z1, HD(6), HD(7), NBATCH, DIM);
  gemm16<<<dim3((DIM/2)/64, 1), B256, 0, stream>>>(w_z1, HD(2), HD(3), w_z2,
                                                   NBATCH, DIM, DIM/2, 0);
  bn_relu<<<1, DIM/2, 0, stream>>>(w_z2, HD(8), HD(9), NBATCH, DIM/2);
  final_k<<<1, NBATCH, 0, stream>>>(w_z2, HD(4), HD(5), out);
}